// EntropyBottleneckSoS_61589831024667
// MI455X (gfx1250) — compile-verified
//
#include <hip/hip_runtime.h>
#include <cmath>

// EntropyBottleneck SoS quantizer + likelihood, MI455X (gfx1250).
// HBM-bound after (1) saturation-windowed SoS (6 hw v_tanh_f32 instead of 60)
// and (2) runtime affine-composition fast path when all factors f_i == 0.

typedef __attribute__((ext_vector_type(4))) float v4f;   // native vector: OK for nontemporal builtins

namespace {
constexpr float kBeta    = 10.0f;
constexpr float kExtrema = 10.0f;
constexpr float kLikBnd  = 1e-9f;
constexpr float kWin     = 0.8f;     // |x-b| window: |beta*(x-b)| >= 8 -> tanh == +/-1 in fp32
constexpr int   kNS      = 60;
constexpr int   kN       = 8;
constexpr int   kC       = 192;
constexpr int   kHW      = 1024;     // 32*32
}

__device__ __forceinline__ float fast_tanh(float x) {
#if __has_builtin(__builtin_amdgcn_tanhf)
    return __builtin_amdgcn_tanhf(x);          // v_tanh_f32 (gfx1250)
#elif __has_builtin(__builtin_amdgcn_tanh_f32)
    return __builtin_amdgcn_tanh_f32(x);
#else
    float e = __expf(2.0f * x);
    return (e - 1.0f) / (e + 1.0f);
#endif
}

__device__ __forceinline__ float softplus_f(float x) {
    // exp(-|x|) in (0,1] -> 1+e in (1,2]: plain hardware log is fully accurate here,
    // avoids the non-inlined __ocml_log1p_f32 call.
    return fmaxf(x, 0.0f) + __logf(1.0f + __expf(-fabsf(x)));
}

__global__ __launch_bounds__(256)
void ebs_sos_kernel(const float* __restrict__ x,
                    const float* __restrict__ sos_w,
                    const float* __restrict__ sos_b,
                    const float* __restrict__ m0, const float* __restrict__ c0, const float* __restrict__ f0,
                    const float* __restrict__ m1, const float* __restrict__ c1, const float* __restrict__ f1,
                    const float* __restrict__ m2, const float* __restrict__ c2, const float* __restrict__ f2,
                    const float* __restrict__ m3, const float* __restrict__ c3, const float* __restrict__ f3,
                    const float* __restrict__ m4, const float* __restrict__ c4,
                    float* __restrict__ yhat, float* __restrict__ lik_out)
{
    __shared__ float2 s_wb[kNS + 6];        // {0.5*w_i, beta*b_i}; padded with {0, huge}
    __shared__ float  s_pref[kNS + 1];      // prefix sums of w_i
    __shared__ float  s_b0, s_invD;
    __shared__ float  s_sp0[3], s_c0[3], s_tf0[3];
    __shared__ float  s_spM[3][9];
    __shared__ float  s_cM[3][3], s_tfM[3][3];
    __shared__ float  s_sp4[3], s_c4v;
    __shared__ float  s_a, s_d;             // composed affine (fast path)
    __shared__ int    s_fast;

    const int b  = blockIdx.x;              // b = n*C + c (matches flat NCHW layout)
    const int ch = b % kC;
    const int t  = threadIdx.x;

    // ---- stage per-channel / shared params into LDS ----
    if (t < kNS) {
        s_wb[t] = make_float2(0.5f * sos_w[t], kBeta * sos_b[t]);
    } else if (t < kNS + 6) {
        s_wb[t] = make_float2(0.0f, 3.0e37f);      // pad: tanh(-huge) = -1, hw = 0 -> contributes 0
    } else if (t == 66) {
        s_b0   = sos_b[0];
        s_invD = 59.0f / (sos_b[kNS - 1] - sos_b[0]);  // uniform linspace grid
    } else if (t >= 68 && t < 71) {
        int j = t - 68;
        s_sp0[j] = softplus_f(m0[ch * 3 + j]);
        s_c0[j]  = c0[ch * 3 + j];
        s_tf0[j] = fast_tanh(f0[ch * 3 + j]);
    } else if (t >= 71 && t < 98) {
        int i = t - 71, L = i / 9, j = i % 9;
        const float* pm = (L == 0) ? m1 : ((L == 1) ? m2 : m3);
        s_spM[L][j] = softplus_f(pm[ch * 9 + j]);
    } else if (t >= 98 && t < 107) {
        int i = t - 98, L = i / 3, j = i % 3;
        const float* pc = (L == 0) ? c1 : ((L == 1) ? c2 : c3);
        const float* pf = (L == 0) ? f1 : ((L == 1) ? f2 : f3);
        s_cM[L][j]  = pc[ch * 3 + j];
        s_tfM[L][j] = fast_tanh(pf[ch * 3 + j]);
    } else if (t >= 107 && t < 110) {
        s_sp4[t - 107] = softplus_f(m4[ch * 3 + (t - 107)]);
    } else if (t == 110) {
        s_c4v = c4[ch];
    }
    __syncthreads();

    if (t == 0) {
        // prefix sums of w_i (w_i = 2 * hw_i)
        float P = 0.0f;
        s_pref[0] = 0.0f;
        for (int i = 0; i < kNS; ++i) { P += 2.0f * s_wb[i].x; s_pref[i + 1] = P; }
        // fast path detection: all tanh(f) == 0 -> MLP is purely affine
        bool fast = true;
        for (int j = 0; j < 3; ++j) fast = fast && (s_tf0[j] == 0.0f);
        for (int L = 0; L < 3; ++L)
            for (int j = 0; j < 3; ++j) fast = fast && (s_tfM[L][j] == 0.0f);
        s_fast = fast ? 1 : 0;
        // compose affine chain 1->3->3->3->3->1 into scalar a*v + d
        float w[3], bb[3];
        for (int j = 0; j < 3; ++j) { w[j] = s_sp0[j]; bb[j] = s_c0[j]; }
        for (int L = 0; L < 3; ++L) {
            float nw[3], nb[3];
            for (int j = 0; j < 3; ++j) {
                float aw = 0.0f, ab = s_cM[L][j];
                for (int q = 0; q < 3; ++q) {
                    aw = fmaf(s_spM[L][j * 3 + q], w[q], aw);
                    ab = fmaf(s_spM[L][j * 3 + q], bb[q], ab);
                }
                nw[j] = aw; nb[j] = ab;
            }
            for (int j = 0; j < 3; ++j) { w[j] = nw[j]; bb[j] = nb[j]; }
        }
        float av = 0.0f, dv = s_c4v;
        for (int q = 0; q < 3; ++q) {
            av = fmaf(s_sp4[q], w[q], av);
            dv = fmaf(s_sp4[q], bb[q], dv);
        }
        s_a = av; s_d = dv;
    }
    __syncthreads();

    // ---- 4 elements per thread via 128-bit vectors (block covers one H*W slab) ----
    // Streaming data: read once / written once -> non-temporal cache policy.
    const v4f* x4 = (const v4f*)x;
    const v4f xv = __builtin_nontemporal_load(&x4[(size_t)b * 256 + t]);
    const float xe[4] = { xv.x, xv.y, xv.z, xv.w };

    // SoS staircase with saturation window: only ~5 steps near x are
    // non-saturated in fp32 (|beta*(x-b_i)| >= 8 rounds tanh to +/-1).
    float yq[4];
    #pragma unroll
    for (int k = 0; k < 4; ++k) {
        const float xk = xe[k];
        int iLo = (int)ceilf((xk - kWin - s_b0) * s_invD);
        iLo = iLo < 0 ? 0 : (iLo > kNS ? kNS : iLo);
        const float p = kBeta * xk;
        float acc = s_pref[iLo] - kExtrema;        // saturated-low steps contribute w_i each
        #pragma unroll
        for (int j = 0; j < 6; ++j) {
            const float2 wb = s_wb[iLo + j];
            // 0.5*w*(tanh+1) = fma(hw, tanh, hw)
            acc = fmaf(wb.x, fast_tanh(p - wb.y), acc + wb.x);
        }
        yq[k] = acc;
    }

    float lo4[4], up4[4];
    if (s_fast) {
        // all factors are zero -> logits_cumulative is affine: a*v + d
        const float av = s_a, dv = s_d;
        #pragma unroll
        for (int k = 0; k < 4; ++k) {
            lo4[k] = fmaf(av, yq[k] - 0.5f, dv);
            up4[k] = fmaf(av, yq[k] + 0.5f, dv);
        }
    } else {
        // general per-channel MLP with tanh residuals
        #pragma unroll
        for (int k = 0; k < 4; ++k) {
            const float vl = yq[k] - 0.5f, vu = yq[k] + 0.5f;
            float l[3], u[3];
            #pragma unroll
            for (int j = 0; j < 3; ++j) {
                float lj = fmaf(s_sp0[j], vl, s_c0[j]);
                float uj = fmaf(s_sp0[j], vu, s_c0[j]);
                l[j] = fmaf(s_tf0[j], fast_tanh(lj), lj);
                u[j] = fmaf(s_tf0[j], fast_tanh(uj), uj);
            }
            #pragma unroll
            for (int L = 0; L < 3; ++L) {
                float nl[3], nu[3];
                #pragma unroll
                for (int j = 0; j < 3; ++j) {
                    float al = s_cM[L][j], au = s_cM[L][j];
                    #pragma unroll
                    for (int q = 0; q < 3; ++q) {
                        const float w = s_spM[L][j * 3 + q];
                        al = fmaf(w, l[q], al);
                        au = fmaf(w, u[q], au);
                    }
                    nl[j] = fmaf(s_tfM[L][j], fast_tanh(al), al);
                    nu[j] = fmaf(s_tfM[L][j], fast_tanh(au), au);
                }
                #pragma unroll
                for (int j = 0; j < 3; ++j) { l[j] = nl[j]; u[j] = nu[j]; }
            }
            float lo = s_c4v, up = s_c4v;
            #pragma unroll
            for (int q = 0; q < 3; ++q) {
                lo = fmaf(s_sp4[q], l[q], lo);
                up = fmaf(s_sp4[q], u[q], up);
            }
            lo4[k] = lo; up4[k] = up;
        }
    }

    float lik[4];
    #pragma unroll
    for (int k = 0; k < 4; ++k) {
        const float ssum = lo4[k] + up4[k];
        const float sg = (ssum > 0.0f) ? -1.0f : ((ssum < 0.0f) ? 1.0f : 0.0f);
        // sigmoid(z) = 0.5*tanh(z/2)+0.5 -> stays on the tanh pipe
        const float su = fmaf(0.5f, fast_tanh(0.5f * (sg * up4[k])), 0.5f);
        const float sl = fmaf(0.5f, fast_tanh(0.5f * (sg * lo4[k])), 0.5f);
        lik[k] = fmaxf(fabsf(su - sl), kLikBnd);
    }

    v4f* y4 = (v4f*)yhat;
    v4f* l4 = (v4f*)lik_out;
    const size_t o = (size_t)b * 256 + t;
    const v4f yv = { yq[0], yq[1], yq[2], yq[3] };
    const v4f lv = { lik[0], lik[1], lik[2], lik[3] };
    __builtin_nontemporal_store(yv, &y4[o]);
    __builtin_nontemporal_store(lv, &l4[o]);
}

extern "C" void kernel_launch(void* const* d_in, const int* in_sizes, int n_in,
                              void* d_out, int out_size, void* d_ws, size_t ws_size,
                              hipStream_t stream) {
    // setup_inputs() dict order: x, sos_w, sos_b, m0,c0,f0, m1,c1,f1, m2,c2,f2, m3,c3,f3, m4,c4
    const float* x     = (const float*)d_in[0];
    const float* sos_w = (const float*)d_in[1];
    const float* sos_b = (const float*)d_in[2];
    const float* m0 = (const float*)d_in[3];
    const float* c0 = (const float*)d_in[4];
    const float* f0 = (const float*)d_in[5];
    const float* m1 = (const float*)d_in[6];
    const float* c1 = (const float*)d_in[7];
    const float* f1 = (const float*)d_in[8];
    const float* m2 = (const float*)d_in[9];
    const float* c2 = (const float*)d_in[10];
    const float* f2 = (const float*)d_in[11];
    const float* m3 = (const float*)d_in[12];
    const float* c3 = (const float*)d_in[13];
    const float* f3 = (const float*)d_in[14];
    const float* m4 = (const float*)d_in[15];
    const float* c4 = (const float*)d_in[16];

    float* yhat = (float*)d_out;
    float* lik  = yhat + (size_t)kN * kC * kHW;   // outputs concatenated flat

    dim3 grid(kN * kC);   // one block per (n, c) slab of 1024 elements
    dim3 block(256);
    hipLaunchKernelGGL(ebs_sos_kernel, grid, block, 0, stream,
                       x, sos_w, sos_b,
                       m0, c0, f0, m1, c1, f1, m2, c2, f2, m3, c3, f3, m4, c4,
                       yhat, lik);
}